// AssignClsLabel_85538568667208
// MI455X (gfx1250) — compile-verified
//
#include <hip/hip_runtime.h>
#include <hip/hip_bf16.h>
#include <float.h>

// AssignClsLabel: labels[b,n] = any_a( IoU(anchor[b,n], gt[b,a]) >= 0.5 && a < count[b] )
//
// MI455X reasoning:
//  - Traffic: 32MB anchors + 8MB labels + ~8KB gt -> ~1.7us at 23.3 TB/s.
//    Math: B*N*A = 134M pairs * ~9 VALU ops -> VALU-bound by >10x over memory.
//  - So: minimize ops/pair. clamp = v_med3 (1 op). Division eliminated:
//      iou >= 0.5  <=>  3*inter >= area + ga  <=>  3*inter - ga >= area,
//    so -ga folds into the per-pair FMA and the "any" reduction is a running
//    v_max compared once against area at the end: 4 med3 + 2 sub + 1 mul +
//    1 fma + 1 max = 9 ops/pair.
//  - WMMA inapplicable (clamps are non-linear; no matmul structure). The CDNA5
//    feature that fits is the async global->LDS DMA: stage the per-batch gt
//    boxes with global_load_async_to_lds_b128 + s_wait_asynccnt (verified in
//    round-1 disasm), then broadcast ds_load_b128 reads amortized over the
//    4 anchors each thread owns.
//  - Streaming data marked non-temporal (native clang vector types, since the
//    builtin rejects HIP_vector_type); labels packed 4-wide into one
//    global_store_b128 per thread.

typedef float v4f __attribute__((ext_vector_type(4)));
typedef int   v4i __attribute__((ext_vector_type(4)));

#define BLOCK 256
#define ITEMS 4
#define A_MAX 64

__global__ __launch_bounds__(BLOCK) void assign_cls_label_kernel(
    const float* __restrict__ anchors,    // (B,N,4)
    const float* __restrict__ gt,         // (B,A,4) as (y1,x1,y2,x2)
    const int*   __restrict__ gt_counts,  // (B,1)
    const int*   __restrict__ use_anchor_p,
    int* __restrict__ out,                // (B,N) int32 labels
    int N, int A)
{
    __shared__ v4f sGT[A_MAX];

    const int b   = blockIdx.y;
    const int tid = threadIdx.x;
    const long long base = (long long)blockIdx.x * (BLOCK * ITEMS);
    const long long n0   = base + (long long)tid * ITEMS;   // 4 consecutive anchors

    // ---- Stage this batch's gt boxes into LDS via gfx1250 async DMA ----
    if (tid < A && tid < A_MAX) {
        const v4f* gsrc   = (const v4f*)gt + ((long long)b * A + tid);
        unsigned lds_addr = (unsigned)(size_t)(&sGT[tid]);  // low 32 bits = LDS offset
#if defined(__gfx1250__)
        asm volatile("global_load_async_to_lds_b128 %0, %1, off"
                     :: "v"(lds_addr), "v"(gsrc)
                     : "memory");
#else
        sGT[tid] = *gsrc;
#endif
    }
#if defined(__gfx1250__)
    asm volatile("s_wait_asynccnt 0x0" ::: "memory");
#endif
    __syncthreads();

    const int count      = gt_counts[b];      // uniform per block -> scalar loop bound
    const int use_anchor = use_anchor_p[0];

    float y1[ITEMS], x1[ITEMS], y2[ITEMS], x2[ITEMS], area[ITEMS], acc[ITEMS];

    const v4f* abase = (const v4f*)anchors + (long long)b * N;
#pragma unroll
    for (int k = 0; k < ITEMS; ++k) {
        long long nk = n0 + k;
        long long ci = (nk < N) ? nk : (long long)(N - 1);   // safe clamp; store guarded
        v4f a4 = __builtin_nontemporal_load(abase + ci);     // global_load_b128 th:NT
        if (use_anchor) {
            float yc = a4.x, xc = a4.y, h = a4.z, w = a4.w;
            y1[k] = yc - 0.5f * h;  y2[k] = y1[k] + h;
            x1[k] = xc - 0.5f * w;  x2[k] = x1[k] + w;
            area[k] = h * w;
        } else {
            y1[k] = a4.x; x1[k] = a4.y; y2[k] = a4.z; x2[k] = a4.w;
            area[k] = (y2[k] - y1[k]) * (x2[k] - x1[k]);
        }
        acc[k] = -FLT_MAX;
    }

    // ---- Main loop: 9 VALU ops per (anchor,gt) pair ----
#pragma unroll 2
    for (int g = 0; g < count; ++g) {
        v4f gb = sGT[g];                                     // ds_load_b128 broadcast
        float gy1 = gb.x, gx1 = gb.y, gy2 = gb.z, gx2 = gb.w;
        float nga = -((gy2 - gy1) * (gx2 - gx1));            // -ga, amortized /ITEMS
#pragma unroll
        for (int k = 0; k < ITEMS; ++k) {
            float yy1 = __builtin_amdgcn_fmed3f(y1[k], gy1, gy2); // clamp = 1 op
            float yy2 = __builtin_amdgcn_fmed3f(y2[k], gy1, gy2);
            float xx1 = __builtin_amdgcn_fmed3f(x1[k], gx1, gx2);
            float xx2 = __builtin_amdgcn_fmed3f(x2[k], gx1, gx2);
            float inter = (yy2 - yy1) * (xx2 - xx1);              // >= 0 by clamp
            float t = __builtin_fmaf(3.0f, inter, nga);           // 3*inter - ga
            acc[k] = fmaxf(acc[k], t);                            // "any" via v_max
        }
    }

    // label = (max_g(3*inter - ga) >= area)
    const long long obase = (long long)b * N;
    if (n0 + ITEMS <= N) {
        v4i lab;
        lab.x = (acc[0] >= area[0]) ? 1 : 0;
        lab.y = (acc[1] >= area[1]) ? 1 : 0;
        lab.z = (acc[2] >= area[2]) ? 1 : 0;
        lab.w = (acc[3] >= area[3]) ? 1 : 0;
        __builtin_nontemporal_store(lab, (v4i*)(out + obase + n0)); // b128 th:NT
    } else {
#pragma unroll
        for (int k = 0; k < ITEMS; ++k) {
            long long nk = n0 + k;
            if (nk < N) out[obase + nk] = (acc[k] >= area[k]) ? 1 : 0;
        }
    }
}

extern "C" void kernel_launch(void* const* d_in, const int* in_sizes, int n_in,
                              void* d_out, int out_size, void* d_ws, size_t ws_size,
                              hipStream_t stream) {
    const float* anchors    = (const float*)d_in[0]; // (B,N,4) f32
    const float* gt         = (const float*)d_in[1]; // (B,A,4) f32
    const int*   counts     = (const int*)d_in[2];   // (B,1) i32
    const int*   use_anchor = (const int*)d_in[3];   // scalar i32
    int*         out        = (int*)d_out;           // (B,N,1) i32

    const int B = in_sizes[2];                 // gt_counts has B elements
    const int A = in_sizes[1] / (B * 4);       // 64
    const int N = in_sizes[0] / (B * 4);       // 262144

    dim3 block(BLOCK);
    dim3 grid((unsigned)((N + BLOCK * ITEMS - 1) / (BLOCK * ITEMS)), (unsigned)B);
    assign_cls_label_kernel<<<grid, block, 0, stream>>>(anchors, gt, counts,
                                                        use_anchor, out, N, A);
}